// LocallyConnected1d_6820408066062
// MI455X (gfx1250) — compile-verified
//
#include <hip/hip_runtime.h>
#include <hip/hip_bf16.h>

// LocallyConnected1d on MI455X (gfx1250), fp32 via V_WMMA_F32_16X16X4_F32.
//
// out[b,c,o] = (1/sqrt(CIN)) * sum_{i,k} x[b,i,o*S+k] * w[c,i,o,k]
// Per-o GEMM: C_o(128x64) = X_o(128x512) * W_o^T, kk = i*8+k.
//
// Roofline: 2.15 GFLOP vs 76 MB minimum traffic -> HBM-bound (~3.3 us @ 23.3 TB/s)
// once the matrix pipe is used; VALU-only fp32 would be compute-bound (~21 us).

#define Bn    128
#define CINn  64
#define COUTn 64
#define ODn   256
#define Kn    8
#define Sn    4
#define Ln    1028          // (256-1)*4 + 8
#define KKn   (CINn * Kn)   // 512 flattened reduction length
#define RSQRT_CIN 0.125f    // 1/sqrt(64)

typedef __attribute__((ext_vector_type(2))) float v2f;
typedef __attribute__((ext_vector_type(8))) float v8f;

__global__ __launch_bounds__(256) void
lc1d_wmma_f32(const float* __restrict__ x,   // [B][CIN][L]
              const float* __restrict__ w,   // [COUT][CIN][OD][K]
              float* __restrict__ out)       // [B][COUT][OD]
{
    const int o    = blockIdx.x;        // output position, 0..255
    const int tid  = threadIdx.x;
    const int wave = tid >> 5;          // 0..7 -> M tile (rows b = 16*wave .. +15)
    const int lane = tid & 31;
    const int half = lane >> 4;         // 0/1: selects K pair within a 4-wide K step
    const int r    = lane & 15;         // M (A frag) / N (B,C frags) within tile

    const int b_base = wave * 16;

    v8f acc0 = {}, acc1 = {}, acc2 = {}, acc3 = {};

    // A: x[(b_base+r)][i][o*S + k], flattened kk = 4*ks + 2*half (+j)
    const float* xrow = x + (size_t)(b_base + r) * (CINn * Ln) + (size_t)o * Sn;
    // B: w[c][i][o][k], c = 16*t + r  (c-stride = CIN*OD*K floats)
    const size_t cstride = (size_t)CINn * ODn * Kn;
    const float* wrow = w + (size_t)r * cstride + (size_t)o * Kn;

#pragma unroll 2
    for (int ks = 0; ks < KKn / 4; ++ks) {
        const int kk = ks * 4 + half * 2; // even -> (kk,kk+1) share one tap pair
        const int ci = kk >> 3;           // input channel
        const int kt = kk & 7;            // tap within window

        // One global_load_b64 per lane for each fragment (8B aligned).
        v2f a  = *(const v2f*)(xrow + (size_t)ci * Ln + kt);

        const float* wp = wrow + (size_t)ci * (ODn * Kn) + kt;
        v2f b0 = *(const v2f*)(wp + 0  * cstride);
        v2f b1 = *(const v2f*)(wp + 16 * cstride);
        v2f b2 = *(const v2f*)(wp + 32 * cstride);
        v2f b3 = *(const v2f*)(wp + 48 * cstride);

        // D = A x B + C ; 8 args: (neg_a, A, neg_b, B, c_mod, C, reuse_a, reuse_b)
        acc0 = __builtin_amdgcn_wmma_f32_16x16x4_f32(false, a, false, b0,
                                                     (short)0, acc0, false, false);
        acc1 = __builtin_amdgcn_wmma_f32_16x16x4_f32(false, a, false, b1,
                                                     (short)0, acc1, false, false);
        acc2 = __builtin_amdgcn_wmma_f32_16x16x4_f32(false, a, false, b2,
                                                     (short)0, acc2, false, false);
        acc3 = __builtin_amdgcn_wmma_f32_16x16x4_f32(false, a, false, b3,
                                                     (short)0, acc3, false, false);
    }

    // C/D layout: VGPR v -> M = v + 8*half, N = r. Store out[b][c][o].
    // 4B stores strided by 1KB over c; L2 (192MB) merges lines across the
    // 256 o-workgroups before HBM writeback.
#pragma unroll
    for (int v = 0; v < 8; ++v) {
        const int bb = b_base + v + 8 * half;
        float* orow = out + (size_t)bb * (COUTn * ODn) + o;
        orow[(size_t)(r +  0) * ODn] = acc0[v] * RSQRT_CIN;
        orow[(size_t)(r + 16) * ODn] = acc1[v] * RSQRT_CIN;
        orow[(size_t)(r + 32) * ODn] = acc2[v] * RSQRT_CIN;
        orow[(size_t)(r + 48) * ODn] = acc3[v] * RSQRT_CIN;
    }
}

extern "C" void kernel_launch(void* const* d_in, const int* in_sizes, int n_in,
                              void* d_out, int out_size, void* d_ws, size_t ws_size,
                              hipStream_t stream) {
    const float* x = (const float*)d_in[0];   // (128, 64, 1028) fp32
    const float* w = (const float*)d_in[1];   // (1, 64, 64, 256, 8) fp32
    float* out = (float*)d_out;               // (128, 64, 256) fp32

    lc1d_wmma_f32<<<dim3(ODn), dim3(256), 0, stream>>>(x, w, out);
}